// DynGrowingHNN_17197049053670
// MI455X (gfx1250) — compile-verified
//
#include <hip/hip_runtime.h>
#include <hip/hip_bf16.h>
#include <math.h>

// ---- problem constants (match reference) ----
#define N_NODES  100000
#define N_HEDGES 20000
#define N_EDGES  800000
#define HID      64
// concat width = 2*HID = 128
#define NTILES   (N_NODES / 16)   // 6250 row tiles of 16

typedef float v2f __attribute__((ext_vector_type(2)));
typedef float v8f __attribute__((ext_vector_type(8)));

// fp32 WMMA: D(16x16,f32) = A(16x4,f32) * B(4x16,f32) + C
// 8-arg pattern per probe doc: (neg_a, A, neg_b, B, c_mod, C, reuse_a, reuse_b)
__device__ __forceinline__ v8f wmma_f32(v2f a, v2f b, v8f c) {
    return __builtin_amdgcn_wmma_f32_16x16x4_f32(false, a, false, b, (short)0, c,
                                                 false, false);
}

__device__ __forceinline__ float sigmoidf_(float x) {
    return 1.0f / (1.0f + __expf(-x));
}

// ---------------- zero scratch ----------------
__global__ void k_zero(float* __restrict__ p, int n) {
    int i = blockIdx.x * blockDim.x + threadIdx.x;
    if (i < n) p[i] = 0.0f;
}

// ---------------- degree histograms ----------------
__global__ void k_degrees(const int* __restrict__ node_idx,
                          const int* __restrict__ hedge_idx,
                          const int* __restrict__ attr,
                          float* __restrict__ Dd, float* __restrict__ Bd) {
    int e = blockIdx.x * blockDim.x + threadIdx.x;
    if (e >= N_EDGES) return;
    int t = attr[e];
    atomicAdd(&Dd[t * N_NODES  + node_idx[e]],  1.0f);
    atomicAdd(&Bd[t * N_HEDGES + hedge_idx[e]], 1.0f);
}

// ---------------- xl = x @ [Wc0 | Wc1]  -> [N_NODES,128] ----------------
// one wave per 16-row tile; 8 col-tiles of 16; K=64 in 16 steps of 4.
__global__ __launch_bounds__(128) void k_xw(const float* __restrict__ x,
                                            const float* __restrict__ Wc, // [2,64,64]
                                            float* __restrict__ xl) {
    int wave = blockIdx.x * (blockDim.x >> 5) + (threadIdx.x >> 5);
    if (wave >= NTILES) return;                 // wave-uniform: EXEC stays all-1s
    int lane  = threadIdx.x & 31;
    int lo    = lane & 15;
    int khalf = lane >> 4;                      // 0 -> K0,K1 ; 1 -> K2,K3
    int m0    = wave * 16;
    int mrow  = m0 + lo;

    v2f a[16];
    #pragma unroll
    for (int kb = 0; kb < 16; ++kb) {
        const float* xr = x + mrow * 64 + kb * 4 + khalf * 2;
        a[kb].x = xr[0];
        a[kb].y = xr[1];
    }
    v8f c[8] = {};
    #pragma unroll
    for (int kb = 0; kb < 16; ++kb) {
        int k = kb * 4 + khalf * 2;
        #pragma unroll
        for (int nt = 0; nt < 8; ++nt) {
            int col = nt * 16 + lo;             // concat col: t = col>>6, j = col&63
            int t = col >> 6, j = col & 63;
            v2f b;
            b.x = Wc[t * 4096 + k       * 64 + j];
            b.y = Wc[t * 4096 + (k + 1) * 64 + j];
            c[nt] = wmma_f32(a[kb], b, c[nt]);
        }
    }
    #pragma unroll
    for (int nt = 0; nt < 8; ++nt)
        #pragma unroll
        for (int r = 0; r < 8; ++r)
            xl[(m0 + r + 8 * khalf) * 128 + nt * 16 + lo] = c[nt][r];
}

// ---------------- node -> hyperedge scatter (segment_sum) ----------------
__global__ void k_scat_hedge(const int* __restrict__ node_idx,
                             const int* __restrict__ hedge_idx,
                             const int* __restrict__ attr,
                             const float* __restrict__ xl,
                             float* __restrict__ ef) {
    int tid = blockIdx.x * blockDim.x + threadIdx.x;
    int e = tid >> 1;
    if (e >= N_EDGES) return;
    int half = tid & 1;                          // 32 cols per thread
    int t = attr[e];
    int base = t * 64 + half * 32;
    const float* src = xl + node_idx[e]  * 128 + base;
    float*       dst = ef + hedge_idx[e] * 128 + base;
    #pragma unroll
    for (int j = 0; j < 32; ++j) atomicAdd(&dst[j], src[j]);
}

// ---------------- hyperedge -> node scatter, B^{-1} folded in ----------------
__global__ void k_scat_node(const int* __restrict__ node_idx,
                            const int* __restrict__ hedge_idx,
                            const int* __restrict__ attr,
                            const float* __restrict__ ef,
                            const float* __restrict__ Bd,
                            float* __restrict__ no) {
    int tid = blockIdx.x * blockDim.x + threadIdx.x;
    int e = tid >> 1;
    if (e >= N_EDGES) return;
    int half = tid & 1;
    int t = attr[e];
    int he = hedge_idx[e];
    float bv   = Bd[t * N_HEDGES + he];
    float binv = bv > 0.0f ? 1.0f / bv : 0.0f;
    int base = t * 64 + half * 32;
    const float* src = ef + he * 128 + base;
    float*       dst = no + node_idx[e] * 128 + base;
    #pragma unroll
    for (int j = 0; j < 32; ++j) atomicAdd(&dst[j], src[j] * binv);
}

// ---------------- h = relu( (no*Dinv + b_conv) @ W_mix + b_mix ) ----------------
__global__ __launch_bounds__(128) void k_mix(const float* __restrict__ no,
                                             const float* __restrict__ Dd,
                                             const float* __restrict__ b_conv, // [128]
                                             const float* __restrict__ W_mix,  // [128,64]
                                             const float* __restrict__ b_mix,  // [64]
                                             float* __restrict__ h) {
    int wave = blockIdx.x * (blockDim.x >> 5) + (threadIdx.x >> 5);
    if (wave >= NTILES) return;
    int lane  = threadIdx.x & 31;
    int lo    = lane & 15;
    int khalf = lane >> 4;
    int m0    = wave * 16;
    int mrow  = m0 + lo;

    float d0 = Dd[mrow];              float di0 = d0 > 0.0f ? 1.0f / d0 : 0.0f;
    float d1 = Dd[N_NODES + mrow];    float di1 = d1 > 0.0f ? 1.0f / d1 : 0.0f;

    v2f a[32];
    #pragma unroll
    for (int kb = 0; kb < 32; ++kb) {
        int k = kb * 4 + khalf * 2;
        float e0 = no[mrow * 128 + k];
        float e1 = no[mrow * 128 + k + 1];
        e0 = e0 * ((k       >> 6) ? di1 : di0) + b_conv[k];
        e1 = e1 * (((k + 1) >> 6) ? di1 : di0) + b_conv[k + 1];
        a[kb].x = e0;
        a[kb].y = e1;
    }
    v8f c[4] = {};
    #pragma unroll
    for (int kb = 0; kb < 32; ++kb) {
        int k = kb * 4 + khalf * 2;
        #pragma unroll
        for (int nt = 0; nt < 4; ++nt) {
            int col = nt * 16 + lo;
            v2f b;
            b.x = W_mix[k       * 64 + col];
            b.y = W_mix[(k + 1) * 64 + col];
            c[nt] = wmma_f32(a[kb], b, c[nt]);
        }
    }
    #pragma unroll
    for (int nt = 0; nt < 4; ++nt)
        #pragma unroll
        for (int r = 0; r < 8; ++r) {
            int m   = m0 + r + 8 * khalf;
            int col = nt * 16 + lo;
            float v = c[nt][r] + b_mix[col];
            h[m * 64 + col] = fmaxf(v, 0.0f);
        }
}

// ---------------- fused GRU cell: h_next = GRU(h, h_prev) ----------------
__global__ __launch_bounds__(128) void k_gru(const float* __restrict__ h,
                                             const float* __restrict__ h_prev,
                                             const float* __restrict__ W_ih, // [64,192]
                                             const float* __restrict__ W_hh, // [64,192]
                                             const float* __restrict__ b_ih, // [192]
                                             const float* __restrict__ b_hh, // [192]
                                             float* __restrict__ h_next) {
    int wave = blockIdx.x * (blockDim.x >> 5) + (threadIdx.x >> 5);
    if (wave >= NTILES) return;
    int lane  = threadIdx.x & 31;
    int lo    = lane & 15;
    int khalf = lane >> 4;
    int m0    = wave * 16;
    int mrow  = m0 + lo;

    v2f ah[16], ap[16];
    #pragma unroll
    for (int kb = 0; kb < 16; ++kb) {
        int k = kb * 4 + khalf * 2;
        ah[kb].x = h[mrow * 64 + k];       ah[kb].y = h[mrow * 64 + k + 1];
        ap[kb].x = h_prev[mrow * 64 + k];  ap[kb].y = h_prev[mrow * 64 + k + 1];
    }

    for (int ct = 0; ct < 4; ++ct) {
        v8f gir = {}, giz = {}, gin = {}, ghr = {}, ghz = {}, ghn = {};
        int cr = ct * 16 + lo;          // reset cols  [0,64)
        int cz = 64  + cr;              // update cols [64,128)
        int cn = 128 + cr;              // new cols    [128,192)
        #pragma unroll
        for (int kb = 0; kb < 16; ++kb) {
            int k  = kb * 4 + khalf * 2;
            int r0 = k * 192, r1 = (k + 1) * 192;
            v2f b;
            b.x = W_ih[r0 + cr]; b.y = W_ih[r1 + cr]; gir = wmma_f32(ah[kb], b, gir);
            b.x = W_ih[r0 + cz]; b.y = W_ih[r1 + cz]; giz = wmma_f32(ah[kb], b, giz);
            b.x = W_ih[r0 + cn]; b.y = W_ih[r1 + cn]; gin = wmma_f32(ah[kb], b, gin);
            b.x = W_hh[r0 + cr]; b.y = W_hh[r1 + cr]; ghr = wmma_f32(ap[kb], b, ghr);
            b.x = W_hh[r0 + cz]; b.y = W_hh[r1 + cz]; ghz = wmma_f32(ap[kb], b, ghz);
            b.x = W_hh[r0 + cn]; b.y = W_hh[r1 + cn]; ghn = wmma_f32(ap[kb], b, ghn);
        }
        float bir = b_ih[cr], biz = b_ih[cz], bin = b_ih[cn];
        float bhr = b_hh[cr], bhz = b_hh[cz], bhn = b_hh[cn];
        #pragma unroll
        for (int r = 0; r < 8; ++r) {
            int m   = m0 + r + 8 * khalf;
            int col = ct * 16 + lo;
            float rg = sigmoidf_(gir[r] + bir + ghr[r] + bhr);
            float zg = sigmoidf_(giz[r] + biz + ghz[r] + bhz);
            float ng = tanhf(gin[r] + bin + rg * (ghn[r] + bhn));
            float hp = h_prev[m * 64 + col];
            h_next[m * 64 + col] = (1.0f - zg) * ng + zg * hp;
        }
    }
}

// ---------------- readout: out = (h_next @ W_ro + b_ro)[:, :3] ----------------
__global__ void k_readout(const float* __restrict__ h_next,
                          const float* __restrict__ W_ro, // [64,64]
                          const float* __restrict__ b_ro,
                          float* __restrict__ out3) {
    int m = blockIdx.x * blockDim.x + threadIdx.x;
    if (m >= N_NODES) return;
    float s0 = b_ro[0], s1 = b_ro[1], s2 = b_ro[2];
    for (int k = 0; k < 64; ++k) {
        float hv = h_next[m * 64 + k];
        s0 += hv * W_ro[k * 64 + 0];
        s1 += hv * W_ro[k * 64 + 1];
        s2 += hv * W_ro[k * 64 + 2];
    }
    out3[m * 3 + 0] = s0;
    out3[m * 3 + 1] = s1;
    out3[m * 3 + 2] = s2;
}

extern "C" void kernel_launch(void* const* d_in, const int* in_sizes, int n_in,
                              void* d_out, int out_size, void* d_ws, size_t ws_size,
                              hipStream_t stream) {
    const float* x       = (const float*)d_in[0];
    const float* h_prev  = (const float*)d_in[1];
    const int*   node_ix = (const int*)  d_in[2];
    const int*   hedg_ix = (const int*)  d_in[3];
    const int*   attr    = (const int*)  d_in[4];
    const float* W_conv  = (const float*)d_in[5];
    const float* b_conv  = (const float*)d_in[6];
    const float* W_mix   = (const float*)d_in[7];
    const float* b_mix   = (const float*)d_in[8];
    const float* W_ih    = (const float*)d_in[9];
    const float* W_hh    = (const float*)d_in[10];
    const float* b_ih    = (const float*)d_in[11];
    const float* b_hh    = (const float*)d_in[12];
    const float* W_ro    = (const float*)d_in[13];
    const float* b_ro    = (const float*)d_in[14];

    // workspace layout (floats): [xl 12.8M][ef 2.56M][no 12.8M][D 0.2M][B 0.04M]
    float* ws  = (float*)d_ws;
    float* xl  = ws;
    float* ef  = xl + (size_t)N_NODES * 128;
    float* no  = ef + (size_t)N_HEDGES * 128;
    float* Dd  = no + (size_t)N_NODES * 128;
    float* Bd  = Dd + (size_t)2 * N_NODES;
    float* h   = xl;                         // xl is dead after k_scat_hedge; reuse

    float* h_next = (float*)d_out;                       // [N_NODES,64]
    float* out3   = h_next + (size_t)N_NODES * 64;       // [N_NODES,3]

    const int ZN = N_HEDGES * 128 + N_NODES * 128 + 2 * N_NODES + 2 * N_HEDGES;
    k_zero<<<(ZN + 255) / 256, 256, 0, stream>>>(ef, ZN);

    k_degrees<<<(N_EDGES + 255) / 256, 256, 0, stream>>>(node_ix, hedg_ix, attr, Dd, Bd);

    k_xw<<<(NTILES + 3) / 4, 128, 0, stream>>>(x, W_conv, xl);

    k_scat_hedge<<<(2 * N_EDGES + 255) / 256, 256, 0, stream>>>(node_ix, hedg_ix, attr,
                                                                xl, ef);
    k_scat_node<<<(2 * N_EDGES + 255) / 256, 256, 0, stream>>>(node_ix, hedg_ix, attr,
                                                               ef, Bd, no);

    k_mix<<<(NTILES + 3) / 4, 128, 0, stream>>>(no, Dd, b_conv, W_mix, b_mix, h);

    k_gru<<<(NTILES + 3) / 4, 128, 0, stream>>>(h, h_prev, W_ih, W_hh, b_ih, b_hh,
                                                h_next);

    k_readout<<<(N_NODES + 255) / 256, 256, 0, stream>>>(h_next, W_ro, b_ro, out3);
}